// MANO_44384192037095
// MI455X (gfx1250) — compile-verified
//
#include <hip/hip_runtime.h>
#include <math.h>

// ---------------- problem constants ----------------
#define NVERT 778
#define NJOINT 16
#define NBETA 10
#define NPOSE 135
#define BATCH 8192
#define XK 148          // padded K: 10 (beta) + 135 (pose_feat) -> 148 (mult of 4)
#define NCOL 2334       // NVERT*3
#define LDN 2336        // padded N (mult of 16)

typedef __attribute__((ext_vector_type(2))) float v2f;
typedef __attribute__((ext_vector_type(8))) float v8f;

#define AS1 __attribute__((address_space(1)))
#define AS3 __attribute__((address_space(3)))

// inverse of REMAP for the 16 posed joints: out position of source joint j
__device__ __constant__ int c_inv16[NJOINT] =
    {0, 5, 6, 7, 9, 10, 11, 17, 18, 19, 13, 14, 15, 1, 2, 3};
// fingertip vertex ids and their output joint positions (inverse REMAP of 16..20)
__device__ __constant__ int c_ftip[5]     = {745, 333, 444, 555, 672};
__device__ __constant__ int c_ftip_out[5] = {4, 8, 12, 16, 20};

// ---------------------------------------------------------------------------
// A0a: build fused GEMM RHS  M[148][2336] = [ S^T (10 x 2334) ; P (135 x 2334) ; 0 ]
// ---------------------------------------------------------------------------
__global__ void __launch_bounds__(256) build_M(const float* __restrict__ S,
                                               const float* __restrict__ P,
                                               float* __restrict__ Mw) {
    int idx = blockIdx.x * blockDim.x + threadIdx.x;
    if (idx >= XK * LDN) return;
    int row = idx / LDN, col = idx % LDN;
    float val = 0.f;
    if (col < NCOL) {
        if (row < NBETA) {
            int v = col / 3, k = col % 3;          // S is [NVERT][3][NBETA]
            val = S[(v * 3 + k) * NBETA + row];
        } else if (row < NBETA + NPOSE) {
            val = P[(row - NBETA) * NCOL + col];   // P is [135][2334]
        }
    }
    Mw[idx] = val;
}

// ---------------------------------------------------------------------------
// A0b: JS[j][k][l] = sum_v Jreg[j,v] S[v,k,l];  JV[j][k] = sum_v Jreg[j,v] V[v,k]
// (collapses the Jrest einsum to a 10-wide dot per joint)
// ---------------------------------------------------------------------------
__global__ void build_JSV(const float* __restrict__ Jreg,
                          const float* __restrict__ S,
                          const float* __restrict__ V,
                          float* __restrict__ JS, float* __restrict__ JV) {
    int t = threadIdx.x;
    if (t < NJOINT * 3 * NBETA) {
        int j = t / (3 * NBETA);
        int rem = t % (3 * NBETA);
        int k = rem / NBETA, l = rem % NBETA;
        float acc = 0.f;
        for (int v = 0; v < NVERT; ++v)
            acc += Jreg[j * NVERT + v] * S[(v * 3 + k) * NBETA + l];
        JS[t] = acc;
    } else if (t < NJOINT * 3 * NBETA + NJOINT * 3) {
        int u = t - NJOINT * 3 * NBETA;
        int j = u / 3, k = u % 3;
        float acc = 0.f;
        for (int v = 0; v < NVERT; ++v)
            acc += Jreg[j * NVERT + v] * V[v * 3 + k];
        JV[u] = acc;
    }
}

// ---------------------------------------------------------------------------
// A1: per-batch Rodrigues + kinematic chain.
//   writes X[b][148]      = [beta | rot[1:]-I | 0,0,0]     (GEMM LHS)
//          Arel[b][16][12] = top-3 rows of A_rel (global skinning transforms)
//          joints_out[b][inv(j)] = posed joint + tvec (16 of the 21 joints)
// ---------------------------------------------------------------------------
__device__ __forceinline__ void rodrigues(float rx, float ry, float rz, float* R) {
    float sq = rx * rx + ry * ry + rz * rz;
    float ang = sqrtf(sq + 1e-8f);
    float inv = 1.0f / ang;
    float x = rx * inv, y = ry * inv, z = rz * inv;
    float s, c;
    sincosf(ang, &s, &c);
    float t = 1.0f - c;
    R[0] = c + t * x * x;     R[1] = t * x * y - s * z; R[2] = t * x * z + s * y;
    R[3] = t * x * y + s * z; R[4] = c + t * y * y;     R[5] = t * y * z - s * x;
    R[6] = t * x * z - s * y; R[7] = t * y * z + s * x; R[8] = c + t * z * z;
}

__global__ void __launch_bounds__(256) pose_chain(
    const float* __restrict__ beta, const float* __restrict__ pose,
    const float* __restrict__ rvec, const float* __restrict__ tvec,
    const float* __restrict__ JS, const float* __restrict__ JV,
    float* __restrict__ X, float* __restrict__ Arel,
    float* __restrict__ joints_out) {
    int b = blockIdx.x * blockDim.x + threadIdx.x;
    if (b >= BATCH) return;

    float bl[NBETA];
#pragma unroll
    for (int l = 0; l < NBETA; ++l) bl[l] = beta[b * NBETA + l];

    float* Xr = X + (size_t)b * XK;
#pragma unroll
    for (int l = 0; l < NBETA; ++l) Xr[l] = bl[l];
    Xr[145] = 0.f; Xr[146] = 0.f; Xr[147] = 0.f;

    // Jrest via collapsed regressor
    float Jr[NJOINT * 3];
#pragma unroll
    for (int j = 0; j < NJOINT; ++j)
#pragma unroll
        for (int k = 0; k < 3; ++k) {
            float acc = JV[j * 3 + k];
#pragma unroll
            for (int l = 0; l < NBETA; ++l) acc += bl[l] * JS[(j * 3 + k) * NBETA + l];
            Jr[j * 3 + k] = acc;
        }

    float tv0 = tvec[b * 3], tv1 = tvec[b * 3 + 1], tv2 = tvec[b * 3 + 2];
    float* Ab = Arel + (size_t)b * (NJOINT * 12);

    auto emit = [&](int j, const float* R, const float* t) {
        const float* Jj = &Jr[j * 3];
        float c0 = R[0] * Jj[0] + R[1] * Jj[1] + R[2] * Jj[2];
        float c1 = R[3] * Jj[0] + R[4] * Jj[1] + R[5] * Jj[2];
        float c2 = R[6] * Jj[0] + R[7] * Jj[1] + R[8] * Jj[2];
        float* o = Ab + j * 12;
        o[0] = R[0]; o[1] = R[1]; o[2]  = R[2]; o[3]  = t[0] - c0;
        o[4] = R[3]; o[5] = R[4]; o[6]  = R[5]; o[7]  = t[1] - c1;
        o[8] = R[6]; o[9] = R[7]; o[10] = R[8]; o[11] = t[2] - c2;
        float* jo = joints_out + ((size_t)b * 21 + c_inv16[j]) * 3;
        jo[0] = t[0] + tv0; jo[1] = t[1] + tv1; jo[2] = t[2] + tv2;
    };

    // root
    float R0[9];
    rodrigues(rvec[b * 3], rvec[b * 3 + 1], rvec[b * 3 + 2], R0);
    float t0[3] = {Jr[0], Jr[1], Jr[2]};
    emit(0, R0, t0);

#pragma unroll
    for (int cc = 0; cc < 5; ++cc) {
        float Rp[9], tp[3];
#pragma unroll
        for (int e = 0; e < 9; ++e) Rp[e] = R0[e];
        tp[0] = t0[0]; tp[1] = t0[1]; tp[2] = t0[2];
#pragma unroll
        for (int i = 0; i < 3; ++i) {
            int j = 1 + cc * 3 + i;
            float Rj[9];
            const float* pr = pose + ((size_t)b * 15 + (j - 1)) * 3;
            rodrigues(pr[0], pr[1], pr[2], Rj);
            // pose_feat = rot - I
#pragma unroll
            for (int e = 0; e < 9; ++e)
                Xr[NBETA + (j - 1) * 9 + e] = Rj[e] - ((e == 0 || e == 4 || e == 8) ? 1.f : 0.f);
            int par = (i == 0) ? 0 : (j - 1);
            float r0 = Jr[j * 3]     - Jr[par * 3];
            float r1 = Jr[j * 3 + 1] - Jr[par * 3 + 1];
            float r2 = Jr[j * 3 + 2] - Jr[par * 3 + 2];
            float Rn[9], tn[3];
#pragma unroll
            for (int m = 0; m < 3; ++m)
#pragma unroll
                for (int n = 0; n < 3; ++n)
                    Rn[m * 3 + n] = Rp[m * 3] * Rj[n] + Rp[m * 3 + 1] * Rj[3 + n] +
                                    Rp[m * 3 + 2] * Rj[6 + n];
            tn[0] = Rp[0] * r0 + Rp[1] * r1 + Rp[2] * r2 + tp[0];
            tn[1] = Rp[3] * r0 + Rp[4] * r1 + Rp[5] * r2 + tp[1];
            tn[2] = Rp[6] * r0 + Rp[7] * r1 + Rp[8] * r2 + tp[2];
            emit(j, Rn, tn);
#pragma unroll
            for (int e = 0; e < 9; ++e) Rp[e] = Rn[e];
            tp[0] = tn[0]; tp[1] = tn[1]; tp[2] = tn[2];
        }
    }
}

// ---------------------------------------------------------------------------
// B: fp32 WMMA GEMM  v_posed[B][2336] = V_bcast + X[B][148] @ M[148][2336]
// Block = 8 waves sharing one 16-column tile of M. The 148x16 M tile (9.25 KB)
// is staged into LDS ONCE per block via GLOBAL_LOAD_ASYNC_TO_LDS_B32
// (ASYNCcnt / s_wait_asynccnt), cutting B-operand VMEM traffic 8x.
// Each wave: one 16(batch) x 16(col) tile, 37 x V_WMMA_F32_16X16X4_F32.
// A layout (ISA 32-bit 16x4): lanes 0-15 hold K={0,1}, lanes 16-31 hold K={2,3}.
// In-loop B reads are ds_load_b32: lanes 0-15 -> banks k*16+0..15,
// lanes 16-31 -> +32 bank set: conflict-free.
// ---------------------------------------------------------------------------
__global__ void __launch_bounds__(256) mano_gemm_wmma(
    const float* __restrict__ X, const float* __restrict__ Mw,
    const float* __restrict__ Vflat, float* __restrict__ vposed) {
    __shared__ float sM[XK * 16];  // M tile, row-major [k][col16]

    const int tid = threadIdx.x;
    const int lane = tid & 31;
    const int wave = tid >> 5;
    const int n0 = blockIdx.x * 16;               // 146 tiles -> cols 0..2335
    const int b0 = (blockIdx.y * 8 + wave) * 16;  // 512 batch tiles
    const int half = lane >> 4;                   // 0: K pair {0,1}, 1: {2,3}
    const int l15 = lane & 15;

    // ---- cooperative async stage of the M tile into LDS ----
    for (int e = tid; e < XK * 16; e += 256) {
        int row = e >> 4, c = e & 15;
        const float* gsrc = Mw + (size_t)row * LDN + (n0 + c);
#if __has_builtin(__builtin_amdgcn_global_load_async_to_lds_b32)
        __builtin_amdgcn_global_load_async_to_lds_b32(
            (AS1 int*)gsrc, (AS3 int*)&sM[e], 0, 0);
#else
        sM[e] = *gsrc;
#endif
    }
#if __has_builtin(__builtin_amdgcn_global_load_async_to_lds_b32)
#if __has_builtin(__builtin_amdgcn_s_wait_asynccnt)
    __builtin_amdgcn_s_wait_asynccnt(0);
#else
    asm volatile("s_wait_asynccnt 0" ::: "memory");
#endif
#endif
    __syncthreads();

    // ---- WMMA K-loop ----
    const float* xrow = X + (size_t)(b0 + l15) * XK + half * 2;
    const int kb = half * 2;

    v8f acc = {};
    for (int k = 0; k < XK; k += 4) {
        v2f a, bb;
        a.x = xrow[k];
        a.y = xrow[k + 1];
        bb.x = sM[(k + kb) * 16 + l15];
        bb.y = sM[(k + kb + 1) * 16 + l15];
        acc = __builtin_amdgcn_wmma_f32_16x16x4_f32(
            false, a, false, bb, (short)0, acc, false, false);
    }
    const int col = n0 + l15;
    const float vadd = (col < NCOL) ? Vflat[col] : 0.f;
#pragma unroll
    for (int r = 0; r < 8; ++r) {
        int brow = b0 + r + half * 8;  // C/D layout: vgpr r, lanes16-31 -> M=r+8
        vposed[(size_t)brow * LDN + col] = acc[r] + vadd;
    }
}

// ---------------------------------------------------------------------------
// C: skinning. Block = one batch x 256 verts; A_rel[b] (192 f32) staged in LDS.
// verts = R_v @ v_posed + t_v + tvec;  fingertips scattered into joints.
// ---------------------------------------------------------------------------
__global__ void __launch_bounds__(256) skin(
    const float* __restrict__ vposed, const float* __restrict__ Arel,
    const float* __restrict__ W, const float* __restrict__ tvec,
    float* __restrict__ verts_out, float* __restrict__ joints_out) {
    __shared__ float sA[NJOINT * 12];
    const int b = blockIdx.y;
    const int tid = threadIdx.x;
    if (tid < NJOINT * 12) sA[tid] = Arel[(size_t)b * (NJOINT * 12) + tid];
    __syncthreads();
    const int v = blockIdx.x * blockDim.x + tid;
    if (v >= NVERT) return;

    float acc[12];
#pragma unroll
    for (int e = 0; e < 12; ++e) acc[e] = 0.f;
#pragma unroll 4
    for (int j = 0; j < NJOINT; ++j) {
        float w = W[v * NJOINT + j];
#pragma unroll
        for (int e = 0; e < 12; ++e) acc[e] += w * sA[j * 12 + e];
    }
    const float* pv = vposed + (size_t)b * LDN + v * 3;
    float p0 = pv[0], p1 = pv[1], p2 = pv[2];
    float tv0 = tvec[b * 3], tv1 = tvec[b * 3 + 1], tv2 = tvec[b * 3 + 2];
    float o0 = acc[0] * p0 + acc[1] * p1 + acc[2]  * p2 + acc[3]  + tv0;
    float o1 = acc[4] * p0 + acc[5] * p1 + acc[6]  * p2 + acc[7]  + tv1;
    float o2 = acc[8] * p0 + acc[9] * p1 + acc[10] * p2 + acc[11] + tv2;
    float* o = verts_out + ((size_t)b * NVERT + v) * 3;
    o[0] = o0; o[1] = o1; o[2] = o2;
#pragma unroll
    for (int f = 0; f < 5; ++f)
        if (v == c_ftip[f]) {
            float* jo = joints_out + ((size_t)b * 21 + c_ftip_out[f]) * 3;
            jo[0] = o0; jo[1] = o1; jo[2] = o2;
        }
}

// ---------------------------------------------------------------------------
extern "C" void kernel_launch(void* const* d_in, const int* in_sizes, int n_in,
                              void* d_out, int out_size, void* d_ws, size_t ws_size,
                              hipStream_t stream) {
    (void)in_sizes; (void)n_in; (void)out_size; (void)ws_size;
    const float* beta = (const float*)d_in[0];
    const float* pose = (const float*)d_in[1];
    const float* rvec = (const float*)d_in[2];
    const float* tvec = (const float*)d_in[3];
    const float* V    = (const float*)d_in[4];
    const float* S    = (const float*)d_in[5];
    const float* P    = (const float*)d_in[6];
    const float* Jreg = (const float*)d_in[7];
    const float* W    = (const float*)d_in[8];

    float* out = (float*)d_out;
    float* verts_out  = out;                              // [B][778][3]
    float* joints_out = out + (size_t)BATCH * NVERT * 3;  // [B][21][3]

    char* ws = (char*)d_ws;
    size_t off = 0;
    float* Mw = (float*)(ws + off); off += (size_t)XK * LDN * 4;
    float* JS = (float*)(ws + off); off += (size_t)NJOINT * 3 * NBETA * 4;
    float* JV = (float*)(ws + off); off += (size_t)NJOINT * 3 * 4;
    off = (off + 255) & ~(size_t)255;
    float* X      = (float*)(ws + off); off += (size_t)BATCH * XK * 4;
    float* Arel   = (float*)(ws + off); off += (size_t)BATCH * NJOINT * 12 * 4;
    float* vposed = (float*)(ws + off); off += (size_t)BATCH * LDN * 4;

    build_M<<<(XK * LDN + 255) / 256, 256, 0, stream>>>(S, P, Mw);
    build_JSV<<<1, 544, 0, stream>>>(Jreg, S, V, JS, JV);
    pose_chain<<<BATCH / 256, 256, 0, stream>>>(beta, pose, rvec, tvec, JS, JV,
                                                X, Arel, joints_out);
    dim3 gB((NCOL + 15) / 16, (BATCH / 16) / 8);  // 146 x 64, 8 waves/block
    mano_gemm_wmma<<<gB, 256, 0, stream>>>(X, Mw, V, vposed);
    dim3 gC((NVERT + 255) / 256, BATCH);
    skin<<<gC, 256, 0, stream>>>(vposed, Arel, W, tvec, verts_out, joints_out);
}